// ContactMap_47691316854895
// MI455X (gfx1250) — compile-verified
//
#include <hip/hip_runtime.h>

typedef __attribute__((ext_vector_type(2))) float v2f;
typedef __attribute__((ext_vector_type(8))) float v8f;

// Problem constants from the reference
constexpr int CB = 256;    // batches
constexpr int CV = 10475;  // vertices
constexpr int CR = 75;     // residues
constexpr int CK = 5;      // points per residue
constexpr int TILES = 25;  // residue-triples per side: 25*3 == 75
constexpr int WAVES_PER_BLOCK = 8;

// One wave computes one 16x16 squared-distance tile (3x3 residue pairs,
// 15 real points + 1 pad per side) with a single V_WMMA_F32_16X16X4_F32:
//   A[m, :] = ( x_i, y_i, z_i, |p_i|^2 )
//   B[:, n] = (-2x_j, -2y_j, -2z_j, 1  )
//   C[m, n] = |q_j|^2
//   D = A*B + C = |p_i|^2 + |q_j|^2 - 2 p_i.q_j   (squared distance)
//
// min-reduction runs on squared distances (sqrt and clamp-at-0 are both
// monotone, so they commute with min); one v_sqrt_f32 per output at the end.
__global__ __launch_bounds__(256)
void ContactMap_wmma_kernel(const float* __restrict__ v1,
                            const float* __restrict__ v2,
                            const long long* __restrict__ rid,
                            float* __restrict__ out)
{
    const int lane = threadIdx.x & 31;
    const int wave = threadIdx.x >> 5;
    const int tile = blockIdx.x * WAVES_PER_BLOCK + wave;

    const int b   = tile / (TILES * TILES);
    const int rem = tile % (TILES * TILES);
    const int ti  = rem / TILES;   // row residue-triple
    const int tj  = rem % TILES;   // col residue-triple

    // Point index inside the 16-wide tile edge (15 == pad; clamped, never read out)
    const int p = lane & 15;
    int resi = ti * 3 + p / 5; if (resi > CR - 1) resi = CR - 1;
    int resj = tj * 3 + p / 5; if (resj > CR - 1) resj = CR - 1;
    const int k = p % 5;

    const long long vidI = rid[resi * CK + k];
    const long long vidJ = rid[resj * CK + k];

    const float* pi = v1 + ((long long)b * CV + vidI) * 3;
    const float* pj = v2 + ((long long)b * CV + vidJ) * 3;
    const float xi = pi[0], yi = pi[1], zi = pi[2];
    const float xj = pj[0], yj = pj[1], zj = pj[2];
    const float rx = xi * xi + yi * yi + zi * zi;
    const float ry = xj * xj + yj * yj + zj * zj;

    // 32-bit 16x4 A layout: lanes 0-15 -> K0,K1 ; lanes 16-31 -> K2,K3.
    const bool hi = lane >= 16;
    v2f a, bmat;
    a.x    = hi ? zi : xi;                     // K0 / K2
    a.y    = hi ? rx : yi;                     // K1 / K3
    bmat.x = hi ? (-2.0f * zj) : (-2.0f * xj); // K0 / K2
    bmat.y = hi ? 1.0f         : (-2.0f * yj); // K1 / K3

    v8f c;
#pragma unroll
    for (int q = 0; q < 8; ++q) c[q] = ry;     // C[m,n] = ry[n] (column-constant)

    // 8 args: (neg_a, A, neg_b, B, c_mod, C, reuse_a, reuse_b)
    v8f dsq = __builtin_amdgcn_wmma_f32_16x16x4_f32(
        /*neg_a=*/false, a, /*neg_b=*/false, bmat,
        /*c_mod=*/(short)0, c, /*reuse_a=*/false, /*reuse_b=*/false);

    // D layout: VGPR q -> row (q or q+8), lane -> column (lane & 15).
    // Reduce squared distances directly.
    const float d0 = dsq[0], d1 = dsq[1], d2 = dsq[2], d3 = dsq[3];
    const float d4 = dsq[4], d5 = dsq[5], d6 = dsq[6], d7 = dsq[7];

    // Row-block partial mins (row blocks: 0-4 / 5-9 / 10-14; row 15 = pad, excluded)
    const float lo04 = fminf(fminf(fminf(d0, d1), fminf(d2, d3)), d4); // rows 0-4   (lo half)
    const float lo57 = fminf(fminf(d5, d6), d7);                      // rows 5-7   (lo half)
    const float hi01 = fminf(d0, d1);                                 // rows 8-9   (hi half)
    const float hi26 = fminf(fminf(fminf(d2, d3), fminf(d4, d5)), d6);// rows 10-14 (hi half)

    // Merge half-waves: lanes 0-15 fetch rows 8-9 and 10-14 from partner lane+16
    const float q89   = __shfl_xor(hi01, 16, 32);
    const float q1014 = __shfl_xor(hi26, 16, 32);
    const float rb0 = lo04;               // min over rows 0-4,  this column
    const float rb1 = fminf(lo57, q89);   // min over rows 5-9
    const float rb2 = q1014;              // min over rows 10-14

    // Column reduce over 5-lane groups (cols 0-4 / 5-9 / 10-14; col 15 = pad)
    const int bj   = lane % 3;            // meaningful for lane < 9
    const int bi   = lane / 3;
    const int base = bj * 5;
    float best = 3.402823466e38f;
#pragma unroll
    for (int cidx = 0; cidx < 5; ++cidx) {
        const float s0 = __shfl(rb0, base + cidx, 32);
        const float s1 = __shfl(rb1, base + cidx, 32);
        const float s2 = __shfl(rb2, base + cidx, 32);
        const float s  = (bi == 0) ? s0 : ((bi == 1) ? s1 : s2);
        best = fminf(best, s);
    }

    // Single clamp + single fast sqrt (v_sqrt_f32) per output.
    const float dist = __builtin_amdgcn_sqrtf(fmaxf(best, 0.0f));

    if (lane < 9) {
        const int Ri = ti * 3 + bi;
        const int Rj = tj * 3 + bj;
        out[((long long)b * CR + Ri) * CR + Rj] = dist;
    }
}

extern "C" void kernel_launch(void* const* d_in, const int* in_sizes, int n_in,
                              void* d_out, int out_size, void* d_ws, size_t ws_size,
                              hipStream_t stream) {
    (void)in_sizes; (void)n_in; (void)out_size; (void)d_ws; (void)ws_size;
    const float*     v1  = (const float*)d_in[0];
    const float*     v2  = (const float*)d_in[1];
    const long long* rid = (const long long*)d_in[2]; // int64 indices
    float*           out = (float*)d_out;

    const int total_tiles = CB * TILES * TILES;           // 160000
    const int blocks = total_tiles / WAVES_PER_BLOCK;     // 20000
    ContactMap_wmma_kernel<<<blocks, WAVES_PER_BLOCK * 32, 0, stream>>>(v1, v2, rid, out);
}